// SiGAT_55430847922810
// MI455X (gfx1250) — compile-verified
//
#include <hip/hip_runtime.h>
#include <math.h>

#define NEG_SLOPE 0.2f

typedef __attribute__((ext_vector_type(2))) float v2f;
typedef __attribute__((ext_vector_type(8))) float v8f;

// ---------------------------------------------------------------------------
// fp32 GEMM via V_WMMA_F32_16X16X4_F32. One wave per 16x16 output tile.
// A: [M,K] row-major, B: [K,Nn] row-major, C: [M,Nn] row-major. M % 16 == 0.
// A-fragment: lane L holds row L%16, K-pair {2*(L/16), 2*(L/16)+1} (b64 load).
// B-fragment: lane L holds col L%16, same K-pair across the 2 VGPRs.
// C/D: VGPR r -> M = r + 8*(L/16), N = L%16.
// ---------------------------------------------------------------------------
__global__ void sigat_gemm_wmma_f32(const float* __restrict__ A,
                                    const float* __restrict__ B,
                                    float* __restrict__ C,
                                    int M, int K, int Nn) {
  const int tilesN = Nn >> 4;
  const int totalTiles = (M >> 4) * tilesN;
  const int wave = (blockIdx.x * blockDim.x + threadIdx.x) >> 5;
  if (wave >= totalTiles) return;              // wave-uniform: EXEC stays all-1s
  const int lane = threadIdx.x & 31;
  const int mt = wave / tilesN;
  const int nt = wave % tilesN;
  const int half = lane >> 4;                  // 0: lanes 0-15, 1: lanes 16-31
  const int l16  = lane & 15;
  const int arow = mt * 16 + l16;              // A-frag row (M)
  const int bcol = nt * 16 + l16;              // B/C-frag col (N)
  const int kk   = half * 2;                   // K-pair base for this lane half

  v8f acc = {};
  const float* __restrict__ arp = A + (size_t)arow * K + kk;
  for (int k0 = 0; k0 < K; k0 += 4) {
    // A pair is contiguous & 8B-aligned -> one global_load_b64.
    v2f a = *reinterpret_cast<const v2f*>(arp + k0);
    v2f b;
    b.x = B[(size_t)(k0 + kk)     * Nn + bcol];
    b.y = B[(size_t)(k0 + kk + 1) * Nn + bcol];
    acc = __builtin_amdgcn_wmma_f32_16x16x4_f32(
        /*neg_a=*/false, a, /*neg_b=*/false, b,
        /*c_mod=*/(short)0, acc, /*reuse_a=*/false, /*reuse_b=*/false);
  }
  const int mbase = mt * 16 + half * 8;
#pragma unroll
  for (int r = 0; r < 8; ++r)
    C[(size_t)(mbase + r) * Nn + bcol] = acc[r];
}

// ---------------------------------------------------------------------------
// a_s[n,h] = sum_c h[n,h,c]*att_src[h,c] ; same for a_d. One thread per (n,h).
// ---------------------------------------------------------------------------
__global__ void sigat_att_scores(const float* __restrict__ h,
                                 const float* __restrict__ atts,
                                 const float* __restrict__ attd,
                                 float* __restrict__ as_, float* __restrict__ ad_,
                                 int n, int H, int C) {
  int i = blockIdx.x * blockDim.x + threadIdx.x;
  if (i >= n * H) return;
  int node = i / H, hd = i % H;
  const float* hp = h + ((size_t)node * H + hd) * C;
  const float* ws = atts + (size_t)hd * C;
  const float* wd = attd + (size_t)hd * C;
  float ss = 0.f, sd = 0.f;
  for (int c = 0; c < C; ++c) { float v = hp[c]; ss += v * ws[c]; sd += v * wd[c]; }
  as_[i] = ss; ad_[i] = sd;
}

// Order-preserving monotonic uint encoding of float (for atomic segment-max).
__device__ __forceinline__ unsigned int sigat_fenc(float f) {
  unsigned int b = __float_as_uint(f);
  return (b & 0x80000000u) ? ~b : (b | 0x80000000u);
}
__device__ __forceinline__ float sigat_fdec(unsigned int e) {
  unsigned int b = (e & 0x80000000u) ? (e & 0x7FFFFFFFu) : ~e;
  return __uint_as_float(b);
}

__device__ __forceinline__ void sigat_edge(int e, int E, const int* __restrict__ src,
                                           const int* __restrict__ dst, int* s, int* d) {
  if (e < E) { *s = src[e]; *d = dst[e]; }
  else       { *s = e - E;  *d = e - E; }     // appended self-loops
}

// Pass 1: segment max of leaky_relu(a_s[src]+a_d[dst]) into encoded m[dst,h].
__global__ void sigat_edge_max(const int* __restrict__ src, const int* __restrict__ dst,
                               const float* __restrict__ as_, const float* __restrict__ ad_,
                               unsigned int* __restrict__ menc, int E, int n, int H) {
  int t = blockIdx.x * blockDim.x + threadIdx.x;
  int total = (E + n) * H;
  if (t >= total) return;
  int e = t / H, hd = t % H, s, d;
  sigat_edge(e, E, src, dst, &s, &d);
  float x = as_[s * H + hd] + ad_[d * H + hd];
  x = x > 0.f ? x : NEG_SLOPE * x;
  atomicMax(&menc[d * H + hd], sigat_fenc(x));
}

// Decode m in place; clamp non-finite (untouched) segments to 0 like reference.
__global__ void sigat_m_finalize(unsigned int* __restrict__ menc, float* __restrict__ m,
                                 int cnt) {
  int i = blockIdx.x * blockDim.x + threadIdx.x;
  if (i >= cnt) return;
  float v = sigat_fdec(menc[i]);
  m[i] = isfinite(v) ? v : 0.f;
}

// Pass 2: denom[dst,h] += exp(e - m[dst,h]).
__global__ void sigat_edge_denom(const int* __restrict__ src, const int* __restrict__ dst,
                                 const float* __restrict__ as_, const float* __restrict__ ad_,
                                 const float* __restrict__ m, float* __restrict__ denom,
                                 int E, int n, int H) {
  int t = blockIdx.x * blockDim.x + threadIdx.x;
  int total = (E + n) * H;
  if (t >= total) return;
  int e = t / H, hd = t % H, s, d;
  sigat_edge(e, E, src, dst, &s, &d);
  float x = as_[s * H + hd] + ad_[d * H + hd];
  x = x > 0.f ? x : NEG_SLOPE * x;
  atomicAdd(&denom[d * H + hd], __expf(x - m[d * H + hd]));
}

// Pass 3: out[dst] += alpha * h[src]. One wave per edge; lane owns PER
// contiguous channels (head index lane-uniform: C/PER lanes per head).
// PER=8 (layer0: H*C=256) -> two float4 loads; PER=2 (layer1: 64) -> one float2.
template <int PER>
__global__ void sigat_edge_scatter(const int* __restrict__ src, const int* __restrict__ dst,
                                   const float* __restrict__ as_, const float* __restrict__ ad_,
                                   const float* __restrict__ m, const float* __restrict__ denom,
                                   const float* __restrict__ h, float* __restrict__ out,
                                   int E, int n, int H, int C) {
  int wave = (blockIdx.x * blockDim.x + threadIdx.x) >> 5;
  int lane = threadIdx.x & 31;
  int total = E + n;
  if (wave >= total) return;
  int s, d;
  sigat_edge(wave, E, src, dst, &s, &d);
  const int HC = PER * 32;                     // H*C
  const int cbase = lane * PER;
  const int hd = cbase / C;                    // constant within a lane
  float x = as_[s * H + hd] + ad_[d * H + hd];
  x = x > 0.f ? x : NEG_SLOPE * x;
  float alpha = __expf(x - m[d * H + hd]) / (denom[d * H + hd] + 1e-16f);
  const float* hs = h + (size_t)s * HC + cbase;
  float* od = out + (size_t)d * HC + cbase;
  if constexpr ((PER & 3) == 0) {
#pragma unroll
    for (int c = 0; c < PER; c += 4) {
      float4 v = *reinterpret_cast<const float4*>(hs + c);   // global_load_b128
      atomicAdd(&od[c + 0], alpha * v.x);
      atomicAdd(&od[c + 1], alpha * v.y);
      atomicAdd(&od[c + 2], alpha * v.z);
      atomicAdd(&od[c + 3], alpha * v.w);
    }
  } else {
#pragma unroll
    for (int c = 0; c < PER; c += 2) {
      float2 v = *reinterpret_cast<const float2*>(hs + c);   // global_load_b64
      atomicAdd(&od[c + 0], alpha * v.x);
      atomicAdd(&od[c + 1], alpha * v.y);
    }
  }
}

// h1 = elu((outp + bp) - (outn + bn))
__global__ void sigat_combine_elu(const float* __restrict__ op, const float* __restrict__ on,
                                  const float* __restrict__ bp, const float* __restrict__ bn,
                                  float* __restrict__ out, int n, int HC) {
  int i = blockIdx.x * blockDim.x + threadIdx.x;
  if (i >= n * HC) return;
  int hc = i % HC;
  float v = (op[i] + bp[hc]) - (on[i] + bn[hc]);
  out[i] = v > 0.f ? v : expm1f(v);
}

// out = (op + bp) - (on + bn)
__global__ void sigat_final_sub(const float* __restrict__ op, const float* __restrict__ on,
                                const float* __restrict__ bp, const float* __restrict__ bn,
                                float* __restrict__ out, int n, int C) {
  int i = blockIdx.x * blockDim.x + threadIdx.x;
  if (i >= n * C) return;
  int c = i % C;
  out[i] = (op[i] + bp[c]) - (on[i] + bn[c]);
}

extern "C" void kernel_launch(void* const* d_in, const int* in_sizes, int n_in,
                              void* d_out, int out_size, void* d_ws, size_t ws_size,
                              hipStream_t stream) {
  (void)n_in; (void)out_size; (void)ws_size;
  const float* x      = (const float*)d_in[0];
  const int*   pei    = (const int*)d_in[1];
  const int*   nei    = (const int*)d_in[2];
  const float* W0p    = (const float*)d_in[3];
  const float* asrc0p = (const float*)d_in[4];
  const float* adst0p = (const float*)d_in[5];
  const float* b0p    = (const float*)d_in[6];
  const float* W0n    = (const float*)d_in[7];
  const float* asrc0n = (const float*)d_in[8];
  const float* adst0n = (const float*)d_in[9];
  const float* b0n    = (const float*)d_in[10];
  const float* W1p    = (const float*)d_in[11];
  const float* asrc1p = (const float*)d_in[12];
  const float* adst1p = (const float*)d_in[13];
  const float* b1p    = (const float*)d_in[14];
  const float* W1n    = (const float*)d_in[15];
  const float* asrc1n = (const float*)d_in[16];
  const float* adst1n = (const float*)d_in[17];
  const float* b1n    = (const float*)d_in[18];

  const int IN_C = 128, HEADS = 4, HID = 64, OUT_C = 64;
  const int HC0 = HEADS * HID;                 // 256
  const int n = in_sizes[0] / IN_C;            // 50000
  const int E = in_sizes[1] / 2;               // 800000
  const int* psrc = pei;     const int* pdst = pei + E;
  const int* nsrc = nei;     const int* ndst = nei + E;

  // Workspace layout (floats): A | B | C | as | ad | m | denom  (~157 MB)
  float* A   = (float*)d_ws;                   // n*256: h (L0), later h1
  float* Bb  = A  + (size_t)n * HC0;           // n*256: out0p, later g (L1)
  float* Cb  = Bb + (size_t)n * HC0;           // n*256: out0n, later out1p|out1n
  float* asb = Cb + (size_t)n * HC0;           // n*HEADS
  float* adb = asb + (size_t)n * HEADS;
  float* mb  = adb + (size_t)n * HEADS;        // also unsigned-int encoded max
  float* dnb = mb  + (size_t)n * HEADS;

  const int totE = E + n;

  auto run_layer = [&](const float* feat, int inC, const float* W,
                       const float* atts, const float* attd,
                       const int* src, const int* dst,
                       int H, int C, float* hbuf, float* outbuf) {
    const int HC = H * C;
    const int waves = (n >> 4) * (HC >> 4);
    sigat_gemm_wmma_f32<<<(waves * 32 + 255) / 256, 256, 0, stream>>>(
        feat, W, hbuf, n, inC, HC);
    sigat_att_scores<<<(n * H + 255) / 256, 256, 0, stream>>>(
        hbuf, atts, attd, asb, adb, n, H, C);
    hipMemsetAsync(mb,  0, (size_t)n * H * sizeof(float), stream);
    hipMemsetAsync(dnb, 0, (size_t)n * H * sizeof(float), stream);
    hipMemsetAsync(outbuf, 0, (size_t)n * HC * sizeof(float), stream);
    sigat_edge_max<<<(totE * H + 255) / 256, 256, 0, stream>>>(
        src, dst, asb, adb, (unsigned int*)mb, E, n, H);
    sigat_m_finalize<<<(n * H + 255) / 256, 256, 0, stream>>>(
        (unsigned int*)mb, mb, n * H);
    sigat_edge_denom<<<(totE * H + 255) / 256, 256, 0, stream>>>(
        src, dst, asb, adb, mb, dnb, E, n, H);
    if (HC == 256) {
      sigat_edge_scatter<8><<<(totE * 32 + 255) / 256, 256, 0, stream>>>(
          src, dst, asb, adb, mb, dnb, hbuf, outbuf, E, n, H, C);
    } else {
      sigat_edge_scatter<2><<<(totE * 32 + 255) / 256, 256, 0, stream>>>(
          src, dst, asb, adb, mb, dnb, hbuf, outbuf, E, n, H, C);
    }
  };

  // Layer 0 (heads=4, concat) for both signs; h -> A, out_p -> Bb, out_n -> Cb.
  run_layer(x, IN_C, W0p, asrc0p, adst0p, psrc, pdst, HEADS, HID, A, Bb);
  run_layer(x, IN_C, W0n, asrc0n, adst0n, nsrc, ndst, HEADS, HID, A, Cb);

  // h1 = elu((out_p + b0p) - (out_n + b0n)) -> A
  sigat_combine_elu<<<(n * HC0 + 255) / 256, 256, 0, stream>>>(
      Bb, Cb, b0p, b0n, A, n, HC0);

  // Layer 1 (heads=1): g -> Bb, out1p -> Cb[0..n*64), out1n -> Cb[n*64..).
  float* out1p = Cb;
  float* out1n = Cb + (size_t)n * OUT_C;
  run_layer(A, HC0, W1p, asrc1p, adst1p, psrc, pdst, 1, OUT_C, Bb, out1p);
  run_layer(A, HC0, W1n, asrc1n, adst1n, nsrc, ndst, 1, OUT_C, Bb, out1n);

  sigat_final_sub<<<(n * OUT_C + 255) / 256, 256, 0, stream>>>(
      out1p, out1n, b1p, b1n, (float*)d_out, n, OUT_C);
}